// QuantumQLSTMCell_65481071409044
// MI455X (gfx1250) — compile-verified
//
#include <hip/hip_runtime.h>
#include <hip/hip_bf16.h>

// ---------------- problem constants (from reference) ----------------
#define TT 512
#define BB 256
#define DD 2048
#define GG 4
#define DQ (DD + 1)        // 2049, linear_w row length
#define ROWS (TT * BB)     // 131072 rows of the projection GEMM
#define KSTEPS (DD / 4)    // 512 K-iterations of wmma_f32_16x16x4

typedef float v2f __attribute__((ext_vector_type(2)));
typedef float v8f __attribute__((ext_vector_type(8)));

// ---------------------------------------------------------------
// Kernel 0: swizzle the gate weights into WMMA-B-fragment-ready form.
// B matrix per K-step is 4x16 (K x N), N=gate (only 0..3 live, rest 0).
// Per ISA layout: VGPR0 holds K=koff, VGPR1 holds K=koff+1 where
// koff = (lane>=16)?2:0, N = lane&15.  We store, for each kstep,
// the float2 each lane needs contiguously: Wtf[kstep*64 + lane*2 + {0,1}].
// Total 512*64 floats = 128 KB, L2-resident during the GEMM.
// ---------------------------------------------------------------
__global__ void qlstm_prep_w(const float* __restrict__ linear_w,
                             float* __restrict__ Wtf) {
    int tid = blockIdx.x * blockDim.x + threadIdx.x;   // 0 .. KSTEPS*32-1
    if (tid >= KSTEPS * 32) return;
    int kstep = tid >> 5;
    int lane  = tid & 31;
    int n     = lane & 15;            // gate column
    int koff  = (lane >> 4) << 1;     // 0 or 2
    int k     = kstep * 4 + koff;
    float w0 = 0.0f, w1 = 0.0f;
    if (n < GG) {
        w0 = linear_w[n * DQ + k];
        w1 = linear_w[n * DQ + k + 1];
    }
    Wtf[kstep * 64 + lane * 2 + 0] = w0;
    Wtf[kstep * 64 + lane * 2 + 1] = w1;
}

// ---------------------------------------------------------------
// Kernel 1: xproj[row, g] = dot(X[row, :], W[g, :DD])  via fp32 WMMA.
// Each wave owns 16 rows; 8 waves / block; 1024 blocks cover 131072 rows.
// A frag: lane (m = lane&15) loads float2 X[row, k+koff .. k+koff+1]
//   -> one global_load_b64 per lane per step, every byte read once.
// B frag: coalesced global_load_b64 from the pre-swizzled Wtf stream.
// Accumulate over 512 steps in v8f, then lanes with N<4 scatter results.
// ---------------------------------------------------------------
__global__ void __launch_bounds__(256)
qlstm_xproj_wmma(const float* __restrict__ X,
                 const float* __restrict__ Wtf,
                 float* __restrict__ xproj) {
    const int lane = threadIdx.x & 31;
    const int wave = threadIdx.x >> 5;
    const int tile = blockIdx.x * 8 + wave;       // 0..8191
    const size_t rowBase = (size_t)tile * 16;

    const int m    = lane & 15;
    const int koff = (lane >> 4) << 1;            // 0 or 2

    const float* __restrict__ a = X + (rowBase + (size_t)m) * DD + koff;
    const float* __restrict__ b = Wtf + lane * 2;

    v8f c = {0.f, 0.f, 0.f, 0.f, 0.f, 0.f, 0.f, 0.f};

#pragma unroll 4
    for (int ks = 0; ks < KSTEPS; ++ks) {
        v2f av = *(const v2f*)a;   a += 4;    // K advances by 4 per step
        v2f bv = *(const v2f*)b;   b += 64;   // next fragment-ready chunk
        // D = A(16x4 f32) * B(4x16 f32) + C ; 8 args:
        // (neg_a, A, neg_b, B, c_mod, C, reuse_a, reuse_b)
        c = __builtin_amdgcn_wmma_f32_16x16x4_f32(
                false, av, false, bv, (short)0, c, false, false);
    }

    // C/D layout: VGPR j -> (lanes 0-15: M=j, N=lane) (lanes 16-31: M=j+8, N=lane-16)
    const int n = lane & 15;
    if (n < GG) {
        const size_t mrow = rowBase + (size_t)((lane >> 4) << 3);
#pragma unroll
        for (int j = 0; j < 8; ++j) {
            xproj[(mrow + (size_t)j) * GG + n] = c[j];
        }
    }
}

// ---------------------------------------------------------------
// Kernel 2: sequential LSTM scan. One thread per batch element (256
// independent recurrences), T=512 steps. Reads xproj as float4 (the 4
// gates), applies the Q=1 "quantum" gate chain, writes outs/h/c.
// d_out layout: outs[T*B] ++ hx[B] ++ cx[B].
// ---------------------------------------------------------------
__global__ void qlstm_scan(const float* __restrict__ xproj,
                           const float* __restrict__ linear_w,
                           const float* __restrict__ linear_b,
                           const float* __restrict__ enc_w,
                           const float* __restrict__ enc_b,
                           const float* __restrict__ rx,
                           const float* __restrict__ meas_w,
                           const float* __restrict__ meas_b,
                           float* __restrict__ out) {
    int bidx = blockIdx.x * blockDim.x + threadIdx.x;
    if (bidx >= BB) return;

    float wh[GG], lbv[GG], ewv[GG], ebv[GG], rxv[GG], mwv[GG], mbv[GG];
#pragma unroll
    for (int g = 0; g < GG; ++g) {
        wh[g]  = linear_w[g * DQ + DD];   // h-weight (last column)
        lbv[g] = linear_b[g];
        ewv[g] = enc_w[g];                // [G,1,1]
        ebv[g] = enc_b[g];
        rxv[g] = rx[g];
        mwv[g] = meas_w[g];               // [G,1,1]
        mbv[g] = meas_b[g];
    }

    float h = 0.0f, cst = 0.0f;
    const float4* __restrict__ xp4 = (const float4*)xproj;

    for (int t = 0; t < TT; ++t) {
        float4 xp = xp4[t * BB + bidx];
        float xv[GG] = {xp.x, xp.y, xp.z, xp.w};
        float vals[GG];
#pragma unroll
        for (int g = 0; g < GG; ++g) {
            float pre = xv[g] + h * wh[g] + lbv[g];
            float ang = pre * ewv[g] + ebv[g];
            float s   = tanhf(rxv[g] * ang);
            vals[g]   = s * mwv[g] + mbv[g];
        }
        float f  = 1.0f / (1.0f + __expf(-vals[0]));
        float i  = 1.0f / (1.0f + __expf(-vals[1]));
        float gg = tanhf(vals[2]);
        float o  = 1.0f / (1.0f + __expf(-vals[3]));
        cst = f * cst + i * gg;
        h   = o * tanhf(cst);
        out[t * BB + bidx] = h;
    }
    out[TT * BB + bidx]      = h;    // hx
    out[TT * BB + BB + bidx] = cst;  // cx
}

// ---------------------------------------------------------------
extern "C" void kernel_launch(void* const* d_in, const int* in_sizes, int n_in,
                              void* d_out, int out_size, void* d_ws, size_t ws_size,
                              hipStream_t stream) {
    const float* inputs   = (const float*)d_in[0];  // [T,B,D]
    const float* linear_w = (const float*)d_in[1];  // [G,1,D+1]
    const float* linear_b = (const float*)d_in[2];  // [G,1]
    const float* enc_w    = (const float*)d_in[3];  // [G,1,1]
    const float* enc_b    = (const float*)d_in[4];  // [G,1]
    const float* rx       = (const float*)d_in[5];  // [G]
    const float* meas_w   = (const float*)d_in[6];  // [G,1,1]
    const float* meas_b   = (const float*)d_in[7];  // [G,1]
    float* out = (float*)d_out;

    // workspace: Wtf (128 KB) then xproj (2 MB)
    float* Wtf   = (float*)d_ws;
    float* xproj = (float*)((char*)d_ws + (size_t)(KSTEPS * 64) * sizeof(float));

    // 0) swizzle weights into B-fragment-ready layout
    qlstm_prep_w<<<(KSTEPS * 32 + 255) / 256, 256, 0, stream>>>(linear_w, Wtf);

    // 1) projection GEMM via fp32 WMMA: 8192 tiles of 16 rows, 8 waves/block
    qlstm_xproj_wmma<<<ROWS / (16 * 8), 256, 0, stream>>>(inputs, Wtf, xproj);

    // 2) sequential LSTM scan, one thread per batch element
    qlstm_scan<<<(BB + 63) / 64, 64, 0, stream>>>(xproj, linear_w, linear_b,
                                                  enc_w, enc_b, rx,
                                                  meas_w, meas_b, out);
}